// RGCNLayer_39719857553723
// MI455X (gfx1250) — compile-verified
//
#include <hip/hip_runtime.h>

typedef __attribute__((ext_vector_type(2))) float v2f;
typedef __attribute__((ext_vector_type(8))) float v8f;

#define N_NODES   100000
#define N_EDGES   600000
#define IN_FEAT   128
#define OUT_FEAT  128
#define NUM_RELS  8
#define NUM_BASES 4
#define LDSTRIDE  132            // 128 + 4 pad: (m*132)%64 = (m*4)%64 -> conflict-free
#define MTILE     64             // rows (edges) per block; 4 WMMA M-tiles per wave
#define PERM_LEN  (N_EDGES + NUM_RELS * MTILE)   // each rel segment padded to x64

// ---------------------------------------------------------------------------
// Build TRANSPOSED weights in workspace (B stored [N][K] so the per-k-step
// fragment {B[k][n], B[k+1][n]} is one aligned b64 load):
//   Wt[r][o][i]  = sum_b w_comp[r,b] * weight[b][i][o]
//   rootT[o][i]  = root_weight[i][o]
// ---------------------------------------------------------------------------
__global__ __launch_bounds__(256) void k_prep_w(const float* __restrict__ weight,
                                                const float* __restrict__ wcomp,
                                                const float* __restrict__ rootw,
                                                float* __restrict__ Wt,
                                                float* __restrict__ rootT) {
    int idx = blockIdx.x * 256 + threadIdx.x;      // exactly (R+1)*128*128 threads
    if (idx < NUM_RELS * 16384) {
        int r = idx >> 14;
        int x = idx & 16383;                       // x = o*128 + i (coalesced write)
        int o = x >> 7;
        int i = x & 127;
        float acc = 0.f;
#pragma unroll
        for (int b = 0; b < NUM_BASES; ++b)
            acc += wcomp[r * NUM_BASES + b] * weight[b * 16384 + i * 128 + o];
        Wt[idx] = acc;
    } else {
        int x = idx - NUM_RELS * 16384;            // < 16384
        int o = x >> 7;
        int i = x & 127;
        rootT[x] = rootw[i * 128 + o];
    }
}

// ---------------------------------------------------------------------------
// bucket-by-relation bookkeeping
// ---------------------------------------------------------------------------
__global__ __launch_bounds__(256) void k_init(int* __restrict__ counts,
                                              int* __restrict__ cursor,
                                              int* __restrict__ perm) {
    int idx = blockIdx.x * 256 + threadIdx.x;
    if (idx < PERM_LEN) perm[idx] = -1;
    if (idx < NUM_RELS) { counts[idx] = 0; cursor[idx] = 0; }
}

__global__ __launch_bounds__(256) void k_count(const int* __restrict__ rel,
                                               int* __restrict__ counts) {
    int e = blockIdx.x * 256 + threadIdx.x;
    if (e < N_EDGES) atomicAdd(&counts[rel[e]], 1);
}

__global__ void k_prefix(const int* __restrict__ counts, int* __restrict__ offsets) {
    int o = 0;
    for (int r = 0; r < NUM_RELS; ++r) {
        offsets[r] = o;
        o += ((counts[r] + MTILE - 1) / MTILE) * MTILE;   // pad segment to 64 edges
    }
    offsets[NUM_RELS] = o;
}

__global__ __launch_bounds__(256) void k_scatter(const int* __restrict__ rel,
                                                 const int* __restrict__ offsets,
                                                 int* __restrict__ cursor,
                                                 int* __restrict__ perm) {
    int e = blockIdx.x * 256 + threadIdx.x;
    if (e < N_EDGES) {
        int r = rel[e];
        int p = atomicAdd(&cursor[r], 1);
        perm[offsets[r] + p] = e;
    }
}

// ---------------------------------------------------------------------------
// out[m0..m0+63, :] = h_tile @ root_weight + bias      (WMMA f32 16x16x4)
// block = 256 threads = 8 waves; wave w owns columns [16w,16w+16), 4 M-tiles.
// B transposed -> one b64 load per k-step; 4 WMMA : 1 VMEM issue ratio.
// ---------------------------------------------------------------------------
__global__ __launch_bounds__(256) void k_root_gemm(const float* __restrict__ h,
                                                   const float* __restrict__ rootT,
                                                   const float* __restrict__ bias,
                                                   float* __restrict__ out) {
    __shared__ float As[MTILE * LDSTRIDE];
    const int m0  = blockIdx.x * MTILE;
    const int tid = threadIdx.x;

    // cooperative load of 64x128 A tile (zero-pad past N_NODES)
#pragma unroll
    for (int j = 0; j < 8; ++j) {
        int idx = tid + j * 256;                   // float4 index, 2048 total
        int r   = idx >> 5;                        // 32 float4 per row
        int c   = (idx & 31) << 2;
        float4 val = make_float4(0.f, 0.f, 0.f, 0.f);
        int row = m0 + r;
        if (row < N_NODES) val = *(const float4*)(h + (size_t)row * IN_FEAT + c);
        *(float4*)(&As[r * LDSTRIDE + c]) = val;
    }
    __syncthreads();

    const int lane = tid & 31;
    const int nb   = (tid >> 5) * 16;
    const int mfr  = lane & 15;
    const int kh   = (lane >> 4) * 2;
    const int n    = nb + (lane & 15);
    const float* Brow = rootT + (size_t)n * IN_FEAT;   // per-lane B row (transposed)

    v8f acc[4] = {};
#pragma unroll 2
    for (int ks = 0; ks < 32; ++ks) {
        int k0 = ks * 4 + kh;
        v2f b = *(const v2f*)(Brow + k0);
#pragma unroll
        for (int t = 0; t < 4; ++t) {
            v2f a;
            a.x = As[(t * 16 + mfr) * LDSTRIDE + k0];
            a.y = As[(t * 16 + mfr) * LDSTRIDE + k0 + 1];
            acc[t] = __builtin_amdgcn_wmma_f32_16x16x4_f32(false, a, false, b,
                                                           (short)0, acc[t], false, false);
        }
    }

    float bv = bias[n];
    if (m0 + MTILE <= N_NODES) {                   // fast path: no row guards
#pragma unroll
        for (int t = 0; t < 4; ++t)
#pragma unroll
            for (int v = 0; v < 8; ++v) {
                int row = m0 + t * 16 + v + 8 * (lane >> 4);
                out[(size_t)row * OUT_FEAT + n] = acc[t][v] + bv;
            }
    } else {
#pragma unroll
        for (int t = 0; t < 4; ++t)
#pragma unroll
            for (int v = 0; v < 8; ++v) {
                int row = m0 + t * 16 + v + 8 * (lane >> 4);
                if (row < N_NODES) out[(size_t)row * OUT_FEAT + n] = acc[t][v] + bv;
            }
    }
}

// ---------------------------------------------------------------------------
// per-relation gather-GEMM-scatter:
//   tile = 64 edges of one relation; A = gathered h[src] rows (LDS);
//   B = Wt[rel] (transposed); C scattered with fp32 atomics into out[dst].
// Padded edges: A row = 0 -> acc contribution exactly 0, dst pinned to node 0
// -> unconditional (branch-free) atomic scatter.
// ---------------------------------------------------------------------------
__global__ __launch_bounds__(256) void k_msg(const float* __restrict__ h,
                                             const float* __restrict__ Wt,
                                             const int* __restrict__ src,
                                             const int* __restrict__ dst,
                                             const int* __restrict__ perm,
                                             const int* __restrict__ offsets,
                                             float* __restrict__ out) {
    __shared__ float As[MTILE * LDSTRIDE];
    __shared__ int   srow[MTILE];
    __shared__ int   drow[MTILE];
    __shared__ int   sh_rel;

    const int tid   = threadIdx.x;
    const int base  = blockIdx.x * MTILE;
    const int total = offsets[NUM_RELS];
    if (base >= total) return;                     // uniform over block

    if (tid == 0) {
        int r = 0;
        while (base >= offsets[r + 1]) ++r;
        sh_rel = r;
    }
    if (tid < MTILE) {
        int e = perm[base + tid];
        srow[tid] = (e >= 0) ? src[e] : -1;        // -1 -> zero A row
        drow[tid] = (e >= 0) ? dst[e] : 0;         // pad -> node 0, adds exact 0.0
    }
    __syncthreads();

    // gather 64x128 A tile from h[src]; padded edges -> zero rows
#pragma unroll
    for (int j = 0; j < 8; ++j) {
        int idx = tid + j * 256;
        int r   = idx >> 5;
        int c   = (idx & 31) << 2;
        int s   = srow[r];
        float4 val = make_float4(0.f, 0.f, 0.f, 0.f);
        if (s >= 0) val = *(const float4*)(h + (size_t)s * IN_FEAT + c);
        *(float4*)(&As[r * LDSTRIDE + c]) = val;
    }
    __syncthreads();

    const int lane = tid & 31;
    const int nb   = (tid >> 5) * 16;
    const int mfr  = lane & 15;
    const int kh   = (lane >> 4) * 2;
    const int n    = nb + (lane & 15);
    const float* Brow = Wt + (size_t)sh_rel * IN_FEAT * OUT_FEAT + (size_t)n * IN_FEAT;

    v8f acc[4] = {};
#pragma unroll 2
    for (int ks = 0; ks < 32; ++ks) {
        int k0 = ks * 4 + kh;
        v2f b = *(const v2f*)(Brow + k0);
#pragma unroll
        for (int t = 0; t < 4; ++t) {
            v2f a;
            a.x = As[(t * 16 + mfr) * LDSTRIDE + k0];
            a.y = As[(t * 16 + mfr) * LDSTRIDE + k0 + 1];
            acc[t] = __builtin_amdgcn_wmma_f32_16x16x4_f32(false, a, false, b,
                                                           (short)0, acc[t], false, false);
        }
    }

    // branch-free scatter: fp32 hardware atomics into L2-resident out
#pragma unroll
    for (int t = 0; t < 4; ++t)
#pragma unroll
        for (int v = 0; v < 8; ++v) {
            int m = t * 16 + v + 8 * (lane >> 4);
            unsafeAtomicAdd(&out[(size_t)drow[m] * OUT_FEAT + n], acc[t][v]);
        }
}

// ---------------------------------------------------------------------------
__global__ __launch_bounds__(256) void k_relu(float* __restrict__ out, int n) {
    int idx = blockIdx.x * 256 + threadIdx.x;
    if (idx < n) out[idx] = fmaxf(out[idx], 0.f);
}

// ---------------------------------------------------------------------------
extern "C" void kernel_launch(void* const* d_in, const int* in_sizes, int n_in,
                              void* d_out, int out_size, void* d_ws, size_t ws_size,
                              hipStream_t stream) {
    (void)in_sizes; (void)n_in; (void)out_size; (void)ws_size;

    const float* h      = (const float*)d_in[0];
    const float* weight = (const float*)d_in[1];
    const float* wcomp  = (const float*)d_in[2];
    const float* rootw  = (const float*)d_in[3];
    const float* bias   = (const float*)d_in[4];
    const int*   src    = (const int*)d_in[5];
    const int*   dst    = (const int*)d_in[6];
    const int*   rel    = (const int*)d_in[7];
    float*       out    = (float*)d_out;

    // ws layout: Wt[8*128*128] | rootT[128*128] | counts[8] | offsets[9] | cursor[8] | perm[]
    float* Wt      = (float*)d_ws;
    float* rootT   = Wt + NUM_RELS * IN_FEAT * OUT_FEAT;
    int*   wsi     = (int*)(rootT + IN_FEAT * OUT_FEAT);
    int*   counts  = wsi;
    int*   offsets = wsi + 8;
    int*   cursor  = wsi + 17;
    int*   perm    = wsi + 25;

    k_prep_w<<<((NUM_RELS + 1) * IN_FEAT * OUT_FEAT) / 256, 256, 0, stream>>>(
        weight, wcomp, rootw, Wt, rootT);
    k_init<<<(PERM_LEN + 255) / 256, 256, 0, stream>>>(counts, cursor, perm);
    k_count<<<(N_EDGES + 255) / 256, 256, 0, stream>>>(rel, counts);
    k_prefix<<<1, 1, 0, stream>>>(counts, offsets);
    k_scatter<<<(N_EDGES + 255) / 256, 256, 0, stream>>>(rel, offsets, cursor, perm);

    k_root_gemm<<<(N_NODES + MTILE - 1) / MTILE, 256, 0, stream>>>(h, rootT, bias, out);

    k_msg<<<PERM_LEN / MTILE, 256, 0, stream>>>(h, Wt, src, dst, perm, offsets, out);

    k_relu<<<(N_NODES * OUT_FEAT + 255) / 256, 256, 0, stream>>>(out, N_NODES * OUT_FEAT);
}